// SignedNet_50818053046717
// MI455X (gfx1250) — compile-verified
//
#include <hip/hip_runtime.h>

// ---------------- problem constants ----------------
#define N_NODES 100000
#define N_EDGES 3200000
#define D_IN    128
#define HID     128
#define H2      256     // 2*HID
#define OUTD    64
#define OUT2    128     // 2*OUT
#define BN_EPS  1e-5f

// ---------------- vector types ----------------
typedef float v2f __attribute__((ext_vector_type(2)));
typedef float v4f __attribute__((ext_vector_type(4)));
typedef float v8f __attribute__((ext_vector_type(8)));

// ---------------- workspace layout (float offsets) ----------------
// invdeg | m (also reused as aggP) | aggN | stats | scale | shift | h
#define OFF_INV   ((size_t)0)
#define OFF_M     ((size_t)100000)
#define OFF_AGGN  (OFF_M     + (size_t)N_NODES * HID)      // 12,900,000
#define OFF_STATS (OFF_AGGN  + (size_t)N_NODES * HID)      // 25,700,000
#define OFF_SCALE (OFF_STATS + 512)
#define OFF_SHIFT (OFF_SCALE + 256)
#define OFF_H     (OFF_SHIFT + 256)                        // 25,701,024
#define ZERO1_N   OFF_H                                    // zero everything before h

// ---------------- helpers ----------------
__device__ __forceinline__ void atomAddF(float* p, float v) {
    // relaxed, device-scope: lowers to hardware global_atomic_add_f32
    __hip_atomic_fetch_add(p, v, __ATOMIC_RELAXED, __HIP_MEMORY_SCOPE_AGENT);
}

__device__ __forceinline__ v8f wmma_f32(v2f a, v2f b, v8f c) {
    // V_WMMA_F32_16X16X4_F32: D = A(16x4) * B(4x16) + C(16x16), fp32 throughout
    return __builtin_amdgcn_wmma_f32_16x16x4_f32(
        /*neg_a=*/false, a, /*neg_b=*/false, b,
        /*c_mod=*/(short)0, c, /*reuse_a=*/false, /*reuse_b=*/false);
}

// ---------------- kernels ----------------

__global__ void zero_kernel(float* __restrict__ p, unsigned n4) {
    unsigned i = blockIdx.x * blockDim.x + threadIdx.x;
    if (i < n4) ((v4f*)p)[i] = (v4f){0.f, 0.f, 0.f, 0.f};
}

__global__ void deg_kernel(const int* __restrict__ dst, float* __restrict__ deg) {
    unsigned e = blockIdx.x * blockDim.x + threadIdx.x;
    if (e < N_EDGES) atomAddF(&deg[dst[e]], 1.0f);
}

__global__ void invdeg_kernel(float* __restrict__ deg) {
    unsigned i = blockIdx.x * blockDim.x + threadIdx.x;
    if (i < N_NODES) {
        float d = deg[i];
        deg[i] = 1.0f / fmaxf(d, 1.0f);   // in-place: becomes inv_deg
    }
}

// one wave (32 lanes) per edge; each lane moves a float4 (128 features total)
__global__ void aggx_kernel(const float* __restrict__ x,
                            const int* __restrict__ src, const int* __restrict__ dst,
                            float* __restrict__ m) {
    unsigned gid = blockIdx.x * blockDim.x + threadIdx.x;
    unsigned e = gid >> 5;
    if (e >= N_EDGES) return;
    unsigned lane = gid & 31u;
    int s = src[e], d = dst[e];
    v4f v = *(const v4f*)(x + (size_t)s * D_IN + lane * 4);
    float* p = m + (size_t)d * D_IN + lane * 4;
    atomAddF(p + 0, v.x); atomAddF(p + 1, v.y);
    atomAddF(p + 2, v.z); atomAddF(p + 3, v.w);
}

// Layer 0: h[:, :128] = (m*invdeg) @ w0_pos_l^T + x @ w0_pos_r^T + b0_pos_r
//          h[:,128:] = x @ w0_neg_r^T + b0_neg_r
// 16 waves/block; wave w owns the 16x16 output tile at columns w*16 of a 16-row stripe.
__global__ __launch_bounds__(512)
void gemm0_kernel(const float* __restrict__ x, const float* __restrict__ m,
                  const float* __restrict__ invdeg,
                  const float* __restrict__ w0pl, const float* __restrict__ w0pr,
                  const float* __restrict__ b0pr, const float* __restrict__ w0nr,
                  const float* __restrict__ b0nr, float* __restrict__ h) {
    const int wave  = threadIdx.x >> 5;     // 0..15
    const int lane  = threadIdx.x & 31;
    const int coll  = lane & 15;
    const int khalf = lane >> 4;            // 0: K=0,1  1: K=2,3
    const int rbase = blockIdx.x * 16;
    const int n0    = wave * 16;            // output column tile in h (0..255)
    const int arow  = rbase + coll;         // A-matrix row held by this lane

    v8f acc = {};
    float bias;
    if (n0 < HID) {
        const float idg = invdeg[arow];
        const float* aM  = m    + (size_t)arow * D_IN + khalf * 2;
        const float* aX  = x    + (size_t)arow * D_IN + khalf * 2;
        const float* bPL = w0pl + (size_t)(n0 + coll) * D_IN + khalf * 2;
        const float* bPR = w0pr + (size_t)(n0 + coll) * D_IN + khalf * 2;
        #pragma unroll 4
        for (int k0 = 0; k0 < D_IN; k0 += 4) {
            v2f am = *(const v2f*)(aM + k0) * idg;   // fold mean scaling
            v2f ax = *(const v2f*)(aX + k0);
            v2f bl = *(const v2f*)(bPL + k0);
            v2f br = *(const v2f*)(bPR + k0);
            acc = wmma_f32(am, bl, acc);
            acc = wmma_f32(ax, br, acc);
        }
        bias = b0pr[n0 + coll];
    } else {
        const int nn = n0 - HID;
        const float* aX  = x    + (size_t)arow * D_IN + khalf * 2;
        const float* bNR = w0nr + (size_t)(nn + coll) * D_IN + khalf * 2;
        #pragma unroll 4
        for (int k0 = 0; k0 < D_IN; k0 += 4) {
            v2f ax = *(const v2f*)(aX + k0);
            v2f bn = *(const v2f*)(bNR + k0);
            acc = wmma_f32(ax, bn, acc);
        }
        bias = b0nr[nn + coll];
    }
    // C layout: VGPR j -> M = j + 8*khalf, N = n0 + coll
    #pragma unroll
    for (int j = 0; j < 8; ++j) {
        int M = j + khalf * 8;
        h[(size_t)(rbase + M) * H2 + n0 + coll] = acc[j] + bias;
    }
}

// BatchNorm stats: each block reduces 250 rows; thread t owns column t.
__global__ __launch_bounds__(256)
void bnstats_kernel(const float* __restrict__ h, float* __restrict__ stats) {
    const int col = threadIdx.x;
    const int r0  = blockIdx.x * 250;
    float s = 0.f, ss = 0.f;
    for (int r = 0; r < 250; ++r) {
        float v = h[(size_t)(r0 + r) * H2 + col];
        s += v; ss += v * v;
    }
    atomAddF(&stats[col], s);
    atomAddF(&stats[H2 + col], ss);
}

__global__ void bnfinal_kernel(const float* __restrict__ stats,
                               const float* __restrict__ gamma, const float* __restrict__ beta,
                               float* __restrict__ scale, float* __restrict__ shift) {
    int c = threadIdx.x;
    if (c < H2) {
        float invn = 1.0f / (float)N_NODES;
        float mu   = stats[c] * invn;
        float var  = stats[H2 + c] * invn - mu * mu;
        float sc   = gamma[c] * rsqrtf(var + BN_EPS);
        scale[c] = sc;
        shift[c] = beta[c] - mu * sc;
    }
}

__global__ void bnapply_kernel(float* __restrict__ h,
                               const float* __restrict__ scale, const float* __restrict__ shift) {
    unsigned i = blockIdx.x * blockDim.x + threadIdx.x;   // over N*64 float4s
    if (i >= (unsigned)N_NODES * (H2 / 4)) return;
    int cb = (i & 63u) * 4;
    v4f v = ((v4f*)h)[i];
    v.x = fmaxf(v.x * scale[cb + 0] + shift[cb + 0], 0.f);
    v.y = fmaxf(v.y * scale[cb + 1] + shift[cb + 1], 0.f);
    v.z = fmaxf(v.z * scale[cb + 2] + shift[cb + 2], 0.f);
    v.w = fmaxf(v.w * scale[cb + 3] + shift[cb + 3], 0.f);
    ((v4f*)h)[i] = v;
}

// one wave per edge; scatter both halves of normalized h (8 atomics/lane)
__global__ void aggh_kernel(const float* __restrict__ h,
                            const int* __restrict__ src, const int* __restrict__ dst,
                            float* __restrict__ aggP, float* __restrict__ aggN) {
    unsigned gid = blockIdx.x * blockDim.x + threadIdx.x;
    unsigned e = gid >> 5;
    if (e >= N_EDGES) return;
    unsigned lane = gid & 31u;
    int s = src[e], d = dst[e];
    v4f vp = *(const v4f*)(h + (size_t)s * H2 + lane * 4);
    v4f vn = *(const v4f*)(h + (size_t)s * H2 + HID + lane * 4);
    float* pp = aggP + (size_t)d * HID + lane * 4;
    float* pn = aggN + (size_t)d * HID + lane * 4;
    atomAddF(pp + 0, vp.x); atomAddF(pp + 1, vp.y);
    atomAddF(pp + 2, vp.z); atomAddF(pp + 3, vp.w);
    atomAddF(pn + 0, vn.x); atomAddF(pn + 1, vn.y);
    atomAddF(pn + 2, vn.z); atomAddF(pn + 3, vn.w);
}

// Layer 1: out[:, :64] = (aggP*invdeg) @ w1_pos_l^T + h @ w1_pos_r^T + b1_pos_r
//          out[:,64:] = (aggN*invdeg) @ w1_neg_l^T + h @ w1_neg_r^T + b1_neg_r
__global__ __launch_bounds__(256)
void gemm1_kernel(const float* __restrict__ h, const float* __restrict__ aggP,
                  const float* __restrict__ aggN, const float* __restrict__ invdeg,
                  const float* __restrict__ w1pl, const float* __restrict__ w1pr,
                  const float* __restrict__ b1pr, const float* __restrict__ w1nl,
                  const float* __restrict__ w1nr, const float* __restrict__ b1nr,
                  float* __restrict__ out) {
    const int wave  = threadIdx.x >> 5;     // 0..7
    const int lane  = threadIdx.x & 31;
    const int coll  = lane & 15;
    const int khalf = lane >> 4;
    const int rbase = blockIdx.x * 16;
    const int n0    = wave * 16;            // output column tile (0..127)
    const int arow  = rbase + coll;

    const bool pos = (n0 < OUTD);
    const float* agg = pos ? aggP : aggN;
    const float* wl  = pos ? w1pl : w1nl;   // (64 x 128) row-major
    const float* wr  = pos ? w1pr : w1nr;   // (64 x 256) row-major
    const float* bb  = pos ? b1pr : b1nr;
    const int nn = pos ? n0 : (n0 - OUTD);

    const float idg = invdeg[arow];
    v8f acc = {};

    // chain 1: mean-aggregated features, K = 128
    {
        const float* aA = agg + (size_t)arow * HID + khalf * 2;
        const float* bL = wl  + (size_t)(nn + coll) * HID + khalf * 2;
        #pragma unroll 4
        for (int k0 = 0; k0 < HID; k0 += 4) {
            v2f a = *(const v2f*)(aA + k0) * idg;
            v2f b = *(const v2f*)(bL + k0);
            acc = wmma_f32(a, b, acc);
        }
    }
    // chain 2: dense h term, K = 256
    {
        const float* aH = h  + (size_t)arow * H2 + khalf * 2;
        const float* bR = wr + (size_t)(nn + coll) * H2 + khalf * 2;
        #pragma unroll 4
        for (int k0 = 0; k0 < H2; k0 += 4) {
            v2f a = *(const v2f*)(aH + k0);
            v2f b = *(const v2f*)(bR + k0);
            acc = wmma_f32(a, b, acc);
        }
    }
    const float bias = bb[nn + coll];
    #pragma unroll
    for (int j = 0; j < 8; ++j) {
        int M = j + khalf * 8;
        out[(size_t)(rbase + M) * OUT2 + n0 + coll] = acc[j] + bias;
    }
}

// ---------------- host launcher ----------------
extern "C" void kernel_launch(void* const* d_in, const int* in_sizes, int n_in,
                              void* d_out, int out_size, void* d_ws, size_t ws_size,
                              hipStream_t stream) {
    (void)in_sizes; (void)n_in; (void)out_size; (void)ws_size;

    const float* x    = (const float*)d_in[0];
    const int*   ei   = (const int*)d_in[1];       // JAX default x64-off -> int32
    const int*   src  = ei;
    const int*   dst  = ei + N_EDGES;
    const float* w0pl = (const float*)d_in[2];
    const float* w0pr = (const float*)d_in[3];
    const float* b0pr = (const float*)d_in[4];
    const float* w0nr = (const float*)d_in[5];
    const float* b0nr = (const float*)d_in[6];
    const float* gam  = (const float*)d_in[7];
    const float* bet  = (const float*)d_in[8];
    const float* w1pl = (const float*)d_in[9];
    const float* w1pr = (const float*)d_in[10];
    const float* b1pr = (const float*)d_in[11];
    const float* w1nl = (const float*)d_in[12];
    const float* w1nr = (const float*)d_in[13];
    const float* b1nr = (const float*)d_in[14];

    float* ws    = (float*)d_ws;
    float* inv   = ws + OFF_INV;     // deg -> inv_deg in place
    float* m     = ws + OFF_M;       // agg(x) accumulator, later reused as aggP
    float* aggN  = ws + OFF_AGGN;
    float* stats = ws + OFF_STATS;
    float* scale = ws + OFF_SCALE;
    float* shift = ws + OFF_SHIFT;
    float* h     = ws + OFF_H;

    // 1. zero invdeg + m + aggN + stats + scale + shift
    {
        unsigned n4 = (unsigned)(ZERO1_N / 4);
        zero_kernel<<<(n4 + 255) / 256, 256, 0, stream>>>(ws, n4);
    }
    // 2. in-degree counts
    deg_kernel<<<(N_EDGES + 255) / 256, 256, 0, stream>>>(dst, inv);
    // 3. inv_deg = 1/max(deg,1)
    invdeg_kernel<<<(N_NODES + 255) / 256, 256, 0, stream>>>(inv);
    // 4. m = segment_sum(x[src], dst)
    aggx_kernel<<<(unsigned)N_EDGES * 32 / 256, 256, 0, stream>>>(x, src, dst, m);
    // 5. layer-0 GEMMs (WMMA f32), writes h (N x 256)
    gemm0_kernel<<<N_NODES / 16, 512, 0, stream>>>(x, m, inv, w0pl, w0pr, b0pr, w0nr, b0nr, h);
    // 6. BN stats
    bnstats_kernel<<<N_NODES / 250, 256, 0, stream>>>(h, stats);
    // 7. BN finalize
    bnfinal_kernel<<<1, 256, 0, stream>>>(stats, gam, bet, scale, shift);
    // 8. BN apply + ReLU in place
    bnapply_kernel<<<((unsigned)N_NODES * (H2 / 4) + 255) / 256, 256, 0, stream>>>(h, scale, shift);
    // 9. re-zero m for reuse as aggP
    {
        unsigned n4 = (unsigned)N_NODES * HID / 4;
        zero_kernel<<<(n4 + 255) / 256, 256, 0, stream>>>(m, n4);
    }
    // 10. scatter both halves of h
    aggh_kernel<<<(unsigned)N_EDGES * 32 / 256, 256, 0, stream>>>(h, src, dst, m, aggN);
    // 11. layer-1 GEMMs (WMMA f32) -> d_out (N x 128)
    gemm1_kernel<<<N_NODES / 16, 256, 0, stream>>>(h, m, aggN, inv,
                                                   w1pl, w1pr, b1pr, w1nl, w1nr, b1nr,
                                                   (float*)d_out);
}